// SpatialEmbLoss_1202590843306
// MI455X (gfx1250) — compile-verified
//
#include <hip/hip_runtime.h>
#include <hip/hip_bf16.h>

// ---------------------------------------------------------------------------
// SpatialEmbLoss for MI455X (gfx1250, wave32).
// Pipeline:
//   k_stats    : TDM (tensor_load_to_lds) mask tiles -> LDS -> WMMA f32 16x16x4
//   k_finstats : centers, s_exp, per-instance var contribution
//   k_emit     : dist / seed-loss partials / packed sort keys (label in LSB)
//   8x (k_hist, k_scan, k_scatter) : stable 4-bit LSD radix sort per instance
//   k_pcnt / k_pscan / k_lovasz    : prefix-of-positives + closed-form dJ dot
//   k_final    : combine scalar loss
// ---------------------------------------------------------------------------

typedef __attribute__((ext_vector_type(2))) float v2f;
typedef __attribute__((ext_vector_type(8))) float v8f;
typedef __attribute__((ext_vector_type(4))) unsigned tdm_g0;
typedef __attribute__((ext_vector_type(8))) int      tdm_g1;
typedef __attribute__((ext_vector_type(4))) int      tdm_g2;

#define BATCH 4
#define MINST 12
#define NI    48            // BATCH*MINST
#define HH    512
#define WW    1024
#define HW    524288        // HH*WW
#define RINV  (1.0f/1023.0f)   // linspace(0,1,1024) step
#define CINV  (1.0f/2047.0f)   // linspace(0,1,2048) step

#define STATS_BLK   256     // blocks per batch in k_stats
#define SB_PIX      2048    // pixels per k_stats block
#define TD_PIX      512     // pixels per TDM tile (12 rows x 512 px x 4B = 24KB)
#define EMIT_TILES  256     // HW / 2048
#define SORT_TILES  128     // HW / 4096

__device__ __forceinline__ float symlogf(float x) {
  return copysignf(log1pf(fabsf(x)), x);
}
__device__ __forceinline__ float rowcoord(int q) { return (float)(q >> 10) * RINV; }
__device__ __forceinline__ float colcoord(int q) { return (float)(q & 1023) * CINV; }

// C/D tile flat index for element (m,n): lane = n + 16*(m>=8), vgpr = m&7
__device__ __host__ __forceinline__ int sidx(int m, int n) {
  int lane = n + ((m >> 3) << 4);
  return lane * 8 + (m & 7);
}

// ---------------------------------------------------------------------------
// Kernel 1: TDM-staged, branchless WMMA mask-weighted reductions.
//   D(16x16) += A(16x4) x B(4x16) over HW in chunks of 4 pixels.
//   A rows = instances (12 used), A cols = 4 pixels of mask (from LDS tile).
//   B rows = 4 pixels, B cols = channels [1, x, y, sigma, sigma^2] (5 used).
//   B operand built as flag-weighted FMA chain: no lane-divergent control flow.
// f32 operand layout: lane l: role=l&15, half=l>>4; v0 -> K=2*half, v1 -> K=2*half+1
// ---------------------------------------------------------------------------
__global__ void k_stats(const float* __restrict__ feats,
                        const int*   __restrict__ masks,
                        float*       __restrict__ partials) {
  const int b    = blockIdx.y;
  const int lane = threadIdx.x & 31;
  const int wave = threadIdx.x >> 5;
  const int m    = lane & 15;         // A row (instance) == B col (channel)
  const int half = lane >> 4;
  const int mrow = (m < MINST) ? m : 0;
  const float mval = (m < MINST) ? 1.0f : 0.0f;

  // per-lane channel flags as floats -> pure FMA combination, no branches
  const float c0f = (m == 0) ? 1.0f : 0.0f;
  const float c1f = (m == 1) ? 1.0f : 0.0f;
  const float c2f = (m == 2) ? 1.0f : 0.0f;
  const float c3f = (m == 3) ? 1.0f : 0.0f;
  const float c4f = (m == 4) ? 1.0f : 0.0f;

  const int blockBase = blockIdx.x * SB_PIX;
  const float* f2 = feats + ((size_t)b * 4 + 2) * HW;

  __shared__ int mtile[MINST * TD_PIX];        // 24KB TDM landing zone
  const unsigned ldsBase = (unsigned)(size_t)(void*)&mtile[0];

  v8f acc = {0.f, 0.f, 0.f, 0.f, 0.f, 0.f, 0.f, 0.f};

  for (int tile = 0; tile < SB_PIX / TD_PIX; ++tile) {
    const int p0 = blockBase + tile * TD_PIX;

    if (wave == 0) {
      // ---- Tensor DMA: 2D tile (12 rows x TD_PIX px, stride HW) -> LDS ----
      const unsigned long long ga =
          (unsigned long long)(const void*)(masks + (size_t)b * MINST * HW + p0);
      tdm_g0 g0;
      g0[0] = 1u;                                           // count=1 (valid D#)
      g0[1] = ldsBase;                                      // lds_addr
      g0[2] = (unsigned)(ga & 0xFFFFFFFFull);               // global_addr lo
      g0[3] = (unsigned)((ga >> 32) & 0x01FFFFFFull) | (2u << 30);  // hi | type=2
      tdm_g1 g1;
      g1[0] = (int)(2u << 16);                              // data_size = 4B
      g1[1] = (int)(((unsigned)HW & 0xFFFFu) << 16);        // tensor_dim0[15:0]
      g1[2] = (int)(((unsigned)HW >> 16) | ((unsigned)MINST << 16)); // dim0 hi | dim1 lo
      g1[3] = (int)((unsigned)TD_PIX << 16);                // dim1 hi=0 | tile_dim0
      g1[4] = (int)MINST;                                   // tile_dim1=12, tile_dim2=0
      g1[5] = (int)HW;                                      // tensor_dim0_stride lo32
      g1[6] = 0;                                            // stride hi | dim1_stride lo
      g1[7] = 0;
      tdm_g2 gz4 = {0, 0, 0, 0};
      tdm_g1 gz8 = {0, 0, 0, 0, 0, 0, 0, 0};
      __builtin_amdgcn_tensor_load_to_lds(g0, g1, gz4, gz4, gz8, 0);
      __builtin_amdgcn_s_wait_tensorcnt(0);
    }
    __syncthreads();

    // ---- 64 pixels per wave, 4 per WMMA, fully branchless ----
    const int wbase = wave * 64;
#pragma unroll 4
    for (int t = 0; t < 16; ++t) {
      const int q0l = wbase + t * 4 + half * 2;     // tile-local, even
      const int g0p = p0 + q0l;                     // global pixel, even

      const int2   mv = *reinterpret_cast<const int2*>(&mtile[mrow * TD_PIX + q0l]);
      const float2 sg = *reinterpret_cast<const float2*>(f2 + g0p);

      v2f av = {mval * (float)mv.x, mval * (float)mv.y};

      const float bx = c0f
                     + c1f * rowcoord(g0p)
                     + c2f * colcoord(g0p)
                     + sg.x * (c3f + c4f * sg.x);
      const float by = c0f
                     + c1f * rowcoord(g0p + 1)
                     + c2f * colcoord(g0p + 1)
                     + sg.y * (c3f + c4f * sg.y);
      v2f bv = {bx, by};

      acc = __builtin_amdgcn_wmma_f32_16x16x4_f32(
          /*neg_a=*/false, av, /*neg_b=*/false, bv,
          /*c_mod=*/(short)0, acc, /*reuse_a=*/false, /*reuse_b=*/false);
    }
    __syncthreads();   // tile consumed before next TDM overwrites it
  }

  __shared__ float red[8][256];
#pragma unroll
  for (int r = 0; r < 8; ++r) red[wave][lane * 8 + r] = acc[r];
  __syncthreads();

  float s = 0.0f;
#pragma unroll
  for (int w = 0; w < 8; ++w) s += red[w][threadIdx.x];
  partials[((size_t)b * STATS_BLK + blockIdx.x) * 256 + threadIdx.x] = s;
}

// ---------------------------------------------------------------------------
// Kernel 2: reduce block partials, derive per-instance params.
// params[inst*8 + {0:cx, 1:cy, 2:sexp, 3:var_sum, 4:msum}]
// ---------------------------------------------------------------------------
__global__ void k_finstats(const float* __restrict__ partials,
                           float*       __restrict__ params) {
  const int b = blockIdx.x;
  const int e = threadIdx.x;
  float sum = 0.0f;
  for (int k = 0; k < STATS_BLK; ++k)
    sum += partials[((size_t)b * STATS_BLK + k) * 256 + e];

  __shared__ float S[256];
  S[e] = sum;
  __syncthreads();

  if (e < MINST) {
    const int m = e;
    const float msum = S[sidx(m, 0)];
    const float sx   = S[sidx(m, 1)];
    const float sy   = S[sidx(m, 2)];
    const float ss   = S[sidx(m, 3)];
    const float ss2  = S[sidx(m, 4)];
    const float denom = msum + 1e-6f;
    const float cx = sx / denom;
    const float cy = sy / denom;
    const float s  = ss / denom;
    const float sexp = expf(10.0f * s);
    const float var  = ss2 - 2.0f * s * ss + (float)HW * s * s;
    float* p = params + (size_t)(b * MINST + m) * 8;
    p[0] = cx; p[1] = cy; p[2] = sexp; p[3] = var; p[4] = msum;
  }
}

// ---------------------------------------------------------------------------
// Kernel 3: dist, seed-loss partials, packed sort keys.
// key: e in [0,2] -> k0 = (bits(e) & ~1) | label ; store u = ~k0
// (ascending u == descending e; ties are order-invariant for the loss)
// ---------------------------------------------------------------------------
__global__ void k_emit(const float* __restrict__ feats,
                       const int*   __restrict__ masks,
                       const float* __restrict__ params,
                       unsigned*    __restrict__ keys,
                       float*       __restrict__ seedPart) {
  const int inst = blockIdx.y;
  const int b = inst / MINST;
  const float cx   = params[(size_t)inst * 8 + 0];
  const float cy   = params[(size_t)inst * 8 + 1];
  const float sexp = params[(size_t)inst * 8 + 2];

  const float* f0 = feats + ((size_t)b * 4 + 0) * HW;
  const float* f1 = feats + ((size_t)b * 4 + 1) * HW;
  const float* f3 = feats + ((size_t)b * 4 + 3) * HW;
  const int*   mp = masks + (size_t)inst * HW;
  unsigned*    kp = keys  + (size_t)inst * HW;

  const int base = blockIdx.x * 2048;
  float seedAcc = 0.0f;

#pragma unroll
  for (int j = 0; j < 8; ++j) {
    const int p = base + threadIdx.x + j * 256;
    __builtin_prefetch(mp + p + 2048, 0, 0);
    const int   mv = mp[p];
    const float sxv = symlogf(f0[p]) + rowcoord(p);
    const float syv = symlogf(f1[p]) + colcoord(p);
    const float dx = sxv - cx, dy = syv - cy;
    const float dist = expf(-(dx * dx + dy * dy) * sexp);
    const float seed = 1.0f / (1.0f + expf(-f3[p]));
    const float t = (seed - dist) * (float)mv;
    seedAcc += t * t;
    float e = 1.0f - (2.0f * dist - 1.0f) * (2.0f * (float)mv - 1.0f);
    e = fmaxf(e, 0.0f);
    const unsigned k0 = (__float_as_uint(e) & ~1u) | (unsigned)(mv & 1);
    kp[p] = ~k0;
  }

  __shared__ float sb[256];
  sb[threadIdx.x] = seedAcc;
  __syncthreads();
  for (int off = 128; off > 0; off >>= 1) {
    if (threadIdx.x < off) sb[threadIdx.x] += sb[threadIdx.x + off];
    __syncthreads();
  }
  if (threadIdx.x == 0)
    seedPart[(size_t)inst * EMIT_TILES + blockIdx.x] = sb[0];
}

// ---------------------------------------------------------------------------
// Radix sort (4 bits/pass, stable). Tile = 4096 keys (256 thr x 16 items).
// hist layout per instance: [digit(16)][tile(128)]
// ---------------------------------------------------------------------------
__global__ void k_hist(const unsigned* __restrict__ src,
                       unsigned*       __restrict__ hist,
                       int shift) {
  const int inst = blockIdx.y, tile = blockIdx.x, t = threadIdx.x;
  __shared__ unsigned h[16];
  if (t < 16) h[t] = 0;
  __syncthreads();
  const size_t base = (size_t)inst * HW + (size_t)tile * 4096;
#pragma unroll
  for (int j = 0; j < 16; ++j) {
    const unsigned u = src[base + t + j * 256];
    atomicAdd(&h[(u >> shift) & 15u], 1u);
  }
  __syncthreads();
  if (t < 16) hist[(size_t)inst * 2048 + (size_t)t * SORT_TILES + tile] = h[t];
}

__global__ void k_scan(unsigned* __restrict__ hist) {
  const int inst = blockIdx.x, t = threadIdx.x;
  unsigned* hp = hist + (size_t)inst * 2048;
  unsigned v[8];
  unsigned run = 0;
#pragma unroll
  for (int j = 0; j < 8; ++j) {
    const unsigned tmp = hp[t * 8 + j];
    v[j] = run;
    run += tmp;
  }
  __shared__ unsigned ls[256];
  ls[t] = run;
  __syncthreads();
  for (int off = 1; off < 256; off <<= 1) {
    const unsigned y = (t >= off) ? ls[t - off] : 0u;
    __syncthreads();
    ls[t] += y;
    __syncthreads();
  }
  const unsigned chunkBase = ls[t] - run;   // exclusive over chunks
#pragma unroll
  for (int j = 0; j < 8; ++j) hp[t * 8 + j] = chunkBase + v[j];
}

__global__ void k_scatter(const unsigned* __restrict__ src,
                          unsigned*       __restrict__ dst,
                          const unsigned* __restrict__ hist,
                          int shift) {
  const int inst = blockIdx.y, tile = blockIdx.x, t = threadIdx.x;
  const size_t base = (size_t)inst * HW + (size_t)tile * 4096;

  unsigned it[16];
#pragma unroll
  for (int j = 0; j < 16; ++j) it[j] = src[base + t * 16 + j];

  // packed per-thread digit counts (8 bits each, counts <= 16)
  unsigned long long c0 = 0ull, c1 = 0ull;
#pragma unroll
  for (int j = 0; j < 16; ++j) {
    const int d = (int)((it[j] >> shift) & 15u);
    if (d < 8) c0 += 1ull << (8 * d); else c1 += 1ull << (8 * (d - 8));
  }

  __shared__ unsigned tstart[16];
  __shared__ unsigned scanbuf[256];
  __shared__ unsigned exclAll[16][256];
  if (t < 16) tstart[t] = hist[(size_t)inst * 2048 + (size_t)t * SORT_TILES + tile];
  __syncthreads();

  for (int d = 0; d < 16; ++d) {
    const unsigned cnt = (d < 8) ? (unsigned)((c0 >> (8 * d)) & 0xFFull)
                                 : (unsigned)((c1 >> (8 * (d - 8))) & 0xFFull);
    scanbuf[t] = cnt;
    __syncthreads();
    for (int off = 1; off < 256; off <<= 1) {
      const unsigned y = (t >= off) ? scanbuf[t - off] : 0u;
      __syncthreads();
      scanbuf[t] += y;
      __syncthreads();
    }
    exclAll[d][t] = scanbuf[t] - cnt;
    __syncthreads();
  }

  unsigned long long r0 = 0ull, r1 = 0ull;
#pragma unroll
  for (int j = 0; j < 16; ++j) {
    const unsigned u = it[j];
    const int d = (int)((u >> shift) & 15u);
    const unsigned run = (d < 8) ? (unsigned)((r0 >> (8 * d)) & 0xFFull)
                                 : (unsigned)((r1 >> (8 * (d - 8))) & 0xFFull);
    const unsigned pos = tstart[d] + exclAll[d][t] + run;
    dst[(size_t)inst * HW + pos] = u;
    if (d < 8) r0 += 1ull << (8 * d); else r1 += 1ull << (8 * (d - 8));
  }
}

// ---------------------------------------------------------------------------
// Lovasz evaluation on sorted keys.
// ---------------------------------------------------------------------------
__global__ void k_pcnt(const unsigned* __restrict__ keys,
                       unsigned*       __restrict__ pcnt) {
  const int inst = blockIdx.y, tile = blockIdx.x, t = threadIdx.x;
  const size_t base = (size_t)inst * HW + (size_t)tile * 4096;
  unsigned c = 0;
#pragma unroll
  for (int j = 0; j < 16; ++j) c += (~keys[base + t + j * 256]) & 1u;
  __shared__ unsigned sb[256];
  sb[t] = c;
  __syncthreads();
  for (int off = 128; off > 0; off >>= 1) {
    if (t < off) sb[t] += sb[t + off];
    __syncthreads();
  }
  if (t == 0) pcnt[(size_t)inst * SORT_TILES + tile] = sb[0];
}

__global__ void k_pscan(unsigned* __restrict__ pcnt,
                        unsigned* __restrict__ ptot) {
  const int inst = blockIdx.x, t = threadIdx.x;   // 128 threads
  unsigned* pp = pcnt + (size_t)inst * SORT_TILES;
  const unsigned v = pp[t];
  __shared__ unsigned ls[128];
  ls[t] = v;
  __syncthreads();
  for (int off = 1; off < 128; off <<= 1) {
    const unsigned y = (t >= off) ? ls[t - off] : 0u;
    __syncthreads();
    ls[t] += y;
    __syncthreads();
  }
  pp[t] = ls[t] - v;                 // exclusive
  if (t == 127) ptot[inst] = ls[t];  // total positives P
}

__global__ void k_lovasz(const unsigned* __restrict__ keys,
                         const unsigned* __restrict__ pcnt,
                         const unsigned* __restrict__ ptot,
                         float*          __restrict__ lov) {
  const int inst = blockIdx.y, tile = blockIdx.x, t = threadIdx.x;
  const size_t base = (size_t)inst * HW + (size_t)tile * 4096;

  unsigned it[16];
  unsigned c = 0;
#pragma unroll
  for (int j = 0; j < 16; ++j) {
    it[j] = keys[base + t * 16 + j];
    c += (~it[j]) & 1u;
  }

  __shared__ unsigned ls[256];
  ls[t] = c;
  __syncthreads();
  for (int off = 1; off < 256; off <<= 1) {
    const unsigned y = (t >= off) ? ls[t - off] : 0u;
    __syncthreads();
    ls[t] += y;
    __syncthreads();
  }
  const unsigned crossExcl = ls[t] - c;

  const unsigned Pu = ptot[inst];
  const float    Pf = (float)Pu;
  unsigned p = pcnt[(size_t)inst * SORT_TILES + tile] + crossExcl;
  float acc = 0.0f;

#pragma unroll
  for (int j = 0; j < 16; ++j) {
    const unsigned k0  = ~it[j];
    const unsigned lab = k0 & 1u;
    const float    e   = __uint_as_float(k0 & ~1u);
    const int i = tile * 4096 + t * 16 + j;   // global sorted index (exclusive)
    const float nf = (float)(i - (int)p);
    float ci;
    if (Pu == 0u) {
      ci = (i == 0) ? e : 0.0f;
    } else if (lab) {
      ci = e / (Pf + nf);
    } else {
      ci = e * (Pf - (float)p) / ((Pf + nf) * (Pf + nf + 1.0f));
    }
    acc += ci;
    p += lab;
  }

  __shared__ float fb[256];
  fb[t] = acc;
  __syncthreads();
  for (int off = 128; off > 0; off >>= 1) {
    if (t < off) fb[t] += fb[t + off];
    __syncthreads();
  }
  if (t == 0) lov[(size_t)inst * SORT_TILES + tile] = fb[0];
}

// ---------------------------------------------------------------------------
// Final combine.
// ---------------------------------------------------------------------------
__global__ void k_final(const float* __restrict__ lov,
                        const float* __restrict__ seedPart,
                        const float* __restrict__ params,
                        float*       __restrict__ out) {
  const int t = threadIdx.x;
  float lv = 0.0f, sd = 0.0f, vr = 0.0f;
  for (int i = t; i < NI * SORT_TILES; i += 256) lv += lov[i];
  for (int i = t; i < NI * EMIT_TILES; i += 256) sd += seedPart[i];
  for (int i = t; i < NI; i += 256) vr += params[(size_t)i * 8 + 3];

  __shared__ float s1[256], s2[256], s3[256];
  s1[t] = lv; s2[t] = sd; s3[t] = vr;
  __syncthreads();
  for (int off = 128; off > 0; off >>= 1) {
    if (t < off) { s1[t] += s1[t + off]; s2[t] += s2[t + off]; s3[t] += s3[t + off]; }
    __syncthreads();
  }
  if (t == 0) {
    const float denom = (float)NI * (float)HW;
    const float inst_loss = s1[0] / (float)NI;
    const float var_loss  = s3[0] / denom;
    const float seed_loss = s2[0] / denom;
    out[0] = inst_loss + 0.01f * var_loss + 0.01f * seed_loss;
  }
}

// ---------------------------------------------------------------------------
// Launch
// ---------------------------------------------------------------------------
extern "C" void kernel_launch(void* const* d_in, const int* in_sizes, int n_in,
                              void* d_out, int out_size, void* d_ws, size_t ws_size,
                              hipStream_t stream) {
  (void)in_sizes; (void)n_in; (void)out_size; (void)ws_size;
  const float* feats = (const float*)d_in[0];
  const int*   masks = (const int*)d_in[1];
  float*       out   = (float*)d_out;
  char*        ws    = (char*)d_ws;

  size_t off = 0;
  auto alloc = [&](size_t bytes) -> char* {
    char* p = ws + off;
    off = (off + bytes + 255) & ~(size_t)255;
    return p;
  };

  float*    partials = (float*)   alloc((size_t)BATCH * STATS_BLK * 256 * 4);
  float*    params   = (float*)   alloc((size_t)NI * 8 * 4);
  float*    seedPart = (float*)   alloc((size_t)NI * EMIT_TILES * 4);
  unsigned* keysA    = (unsigned*)alloc((size_t)NI * HW * 4);
  unsigned* keysB    = (unsigned*)alloc((size_t)NI * HW * 4);
  unsigned* hist     = (unsigned*)alloc((size_t)NI * 2048 * 4);
  unsigned* pcnt     = (unsigned*)alloc((size_t)NI * SORT_TILES * 4);
  unsigned* ptot     = (unsigned*)alloc((size_t)NI * 4);
  float*    lov      = (float*)   alloc((size_t)NI * SORT_TILES * 4);

  k_stats   <<<dim3(STATS_BLK, BATCH), 256, 0, stream>>>(feats, masks, partials);
  k_finstats<<<BATCH, 256, 0, stream>>>(partials, params);
  k_emit    <<<dim3(EMIT_TILES, NI), 256, 0, stream>>>(feats, masks, params, keysA, seedPart);

  unsigned* src = keysA;
  unsigned* dst = keysB;
  for (int pass = 0; pass < 8; ++pass) {
    const int shift = pass * 4;
    k_hist   <<<dim3(SORT_TILES, NI), 256, 0, stream>>>(src, hist, shift);
    k_scan   <<<NI, 256, 0, stream>>>(hist);
    k_scatter<<<dim3(SORT_TILES, NI), 256, 0, stream>>>(src, dst, hist, shift);
    unsigned* tmp = src; src = dst; dst = tmp;
  }
  // after 8 passes, sorted keys are back in keysA (== src)

  k_pcnt  <<<dim3(SORT_TILES, NI), 256, 0, stream>>>(src, pcnt);
  k_pscan <<<NI, 128, 0, stream>>>(pcnt, ptot);
  k_lovasz<<<dim3(SORT_TILES, NI), 256, 0, stream>>>(src, pcnt, ptot, lov);
  k_final <<<1, 256, 0, stream>>>(lov, seedPart, params, out);
}